// SparseMemory_24309514895758
// MI455X (gfx1250) — compile-verified
//
#include <hip/hip_runtime.h>
#include <math.h>

#define DELTA 1e-6f
#define BB 32
#define MM 1024
#define WW_ 64
#define DD 1024
#define RR 4
#define KNN 8

typedef __attribute__((ext_vector_type(2))) float v2f;
typedef __attribute__((ext_vector_type(4))) float v4f;
typedef __attribute__((ext_vector_type(8))) float v8f;

// ---------------- workspace offsets (in floats) ----------------
#define OFF_RK    0        // 32*256 read keys
#define OFF_WK    8192     // 32*64 write key
#define OFF_EV    10240    // 32*64 erase (sigmoid)
#define OFF_WV    12288    // 32*64 write vector
#define OFF_WS    14336    // 32 write strength
#define OFF_FG    14368    // 32*4 free gates (sigmoid)
#define OFF_AG    14496    // 32 alloc gate (sigmoid)
#define OFF_WG    14528    // 32 write gate (sigmoid)
#define OFF_WCW   14560    // 32*1024 content write weights
#define OFF_SUMWW 47328    // 32

// ---------------- output offsets (in floats, concatenated tuple) ----------------
#define O_RV    0          // read_vectors  (32,4,8,64)
#define O_NRW   65536      // new_read_weights (32,4,8)
#define O_MEM   66560      // memory (32,1024,64)
#define O_LINK  2163712    // link (32,1,1024,1024)
#define O_PREC  35718144   // precedence (32,1,1024)
#define O_WW    35750912   // ww (32,1,1024)
#define O_USAGE 35783680   // usage (32,1024)

__device__ __forceinline__ float sigmoidf(float x) {
    return 1.0f / (1.0f + __expf(-x));
}

// ============ WMMA f32 GEMM: out(32,N) = xi(32,1024) @ W(1024,N) + bias ============
// One wave (32 lanes) per 16x16 tile. V_WMMA_F32_16X16X4_F32, K-loop of 256.
__global__ void gemm_wmma_f32(const float* __restrict__ A,   // (32,1024)
                              const float* __restrict__ Wm,  // (1024,N)
                              const float* __restrict__ bias,
                              float* __restrict__ out,       // (32,N)
                              int N, int applySigmoid) {
    const int tn   = blockIdx.x * 16;
    const int tm   = blockIdx.y * 16;
    const int lane = threadIdx.x;       // 0..31, EXEC all ones
    const int half = lane >> 4;         // 0|1
    const int lo   = lane & 15;

    v8f acc = {0.f,0.f,0.f,0.f,0.f,0.f,0.f,0.f};
    const float* arow = A + (size_t)(tm + lo) * DD;

    for (int k = 0; k < DD; k += 4) {
        const int ka = k + 2 * half;           // K pair this lane-half holds
        v2f a;  a.x = arow[ka];               a.y = arow[ka + 1];
        v2f b;  b.x = Wm[(size_t)ka * N + tn + lo];
                b.y = Wm[(size_t)(ka + 1) * N + tn + lo];
        acc = __builtin_amdgcn_wmma_f32_16x16x4_f32(
                  false, a, false, b, (short)0, acc, false, false);
    }
    // C/D layout: acc[v] -> row = v + 8*half, col = lo
    #pragma unroll
    for (int v = 0; v < 8; ++v) {
        const int row = tm + v + 8 * half;
        const int col = tn + lo;
        float val = acc[v] + bias[col];
        if (applySigmoid) val = sigmoidf(val);
        out[(size_t)row * N + col] = val;
    }
}

// ============ tiny projections: write_strength, free_gates, alloc/write gates ======
__global__ void small_proj(const float* __restrict__ xi,
                           const float* __restrict__ W_ws, const float* __restrict__ b_ws,
                           const float* __restrict__ W_fg, const float* __restrict__ b_fg,
                           const float* __restrict__ W_ag, const float* __restrict__ b_ag,
                           const float* __restrict__ W_wg, const float* __restrict__ b_wg,
                           float* __restrict__ ws) {
    int t = threadIdx.x;
    if (t >= BB * 7) return;
    int b = t / 7, c = t % 7;
    const float* x = xi + (size_t)b * DD;
    float s = 0.f;
    if (c == 0) {
        for (int k = 0; k < DD; ++k) s += x[k] * W_ws[k];
        ws[OFF_WS + b] = s + b_ws[0];
    } else if (c <= 4) {
        int r = c - 1;
        for (int k = 0; k < DD; ++k) s += x[k] * W_fg[k * RR + r];
        ws[OFF_FG + b * RR + r] = sigmoidf(s + b_fg[r]);
    } else if (c == 5) {
        for (int k = 0; k < DD; ++k) s += x[k] * W_ag[k];
        ws[OFF_AG + b] = sigmoidf(s + b_ag[0]);
    } else {
        for (int k = 0; k < DD; ++k) s += x[k] * W_wg[k];
        ws[OFF_WG + b] = sigmoidf(s + b_wg[0]);
    }
}

// ============ usage update ============
__global__ void usage_kernel(const float* __restrict__ usage,
                             const float* __restrict__ write_weights, // (b,1,m)
                             const float* __restrict__ read_weights,  // (b,4,m)
                             const float* __restrict__ ws,
                             float* __restrict__ out_usage) {
    int i = blockIdx.x * blockDim.x + threadIdx.x; // 0..32767
    int b = i >> 10, m = i & (MM - 1);
    float u = usage[i];
    u = u + (1.f - u) * write_weights[i];          // prod over size-1 axis
    float psi = 1.f;
    #pragma unroll
    for (int r = 0; r < RR; ++r)
        psi *= 1.f - ws[OFF_FG + b * RR + r] * read_weights[(size_t)b * RR * MM + r * MM + m];
    out_usage[i] = u * psi;
}

// ============ cosine similarity + softmax -> wcw (v4f row loads) ============
__global__ void cos_softmax(const float* __restrict__ memory,
                            const float* __restrict__ ws,
                            float* __restrict__ wswr) {
    __shared__ float sc[MM];
    __shared__ float red[256];
    int b = blockIdx.x, t = threadIdx.x;
    const v4f* key4 = (const v4f*)(ws + OFF_WK + b * WW_);
    float kn = 0.f;
    #pragma unroll
    for (int j = 0; j < WW_ / 4; ++j) {
        v4f k4 = key4[j];
        kn += k4.x * k4.x + k4.y * k4.y + k4.z * k4.z + k4.w * k4.w;
    }
    kn = sqrtf(kn) + DELTA;
    float s = ws[OFF_WS + b];
    for (int m = t; m < MM; m += 256) {
        const v4f* row4 = (const v4f*)(memory + (size_t)b * MM * WW_ + (size_t)m * WW_);
        float dot = 0.f, nn = 0.f;
        #pragma unroll
        for (int j = 0; j < WW_ / 4; ++j) {
            v4f r4 = row4[j], k4 = key4[j];
            dot += r4.x * k4.x + r4.y * k4.y + r4.z * k4.z + r4.w * k4.w;
            nn  += r4.x * r4.x + r4.y * r4.y + r4.z * r4.z + r4.w * r4.w;
        }
        sc[m] = s * (dot / ((sqrtf(nn) + DELTA) * kn));
    }
    __syncthreads();
    float mx = -3.4e38f;
    for (int m = t; m < MM; m += 256) mx = fmaxf(mx, sc[m]);
    red[t] = mx; __syncthreads();
    for (int o = 128; o > 0; o >>= 1) { if (t < o) red[t] = fmaxf(red[t], red[t + o]); __syncthreads(); }
    mx = red[0]; __syncthreads();
    float sum = 0.f;
    for (int m = t; m < MM; m += 256) { float e = __expf(sc[m] - mx); sc[m] = e; sum += e; }
    red[t] = sum; __syncthreads();
    for (int o = 128; o > 0; o >>= 1) { if (t < o) red[t] += red[t + o]; __syncthreads(); }
    sum = red[0];
    for (int m = t; m < MM; m += 256) wswr[b * MM + m] = sc[m] / sum;
}

// ============ bitonic sort + product scan -> allocation -> ww, sum(ww) ============
__global__ void alloc_ww_kernel(const float* __restrict__ usage_new, // d_out O_USAGE
                                const float* __restrict__ wcw,
                                const float* __restrict__ ws,
                                float* __restrict__ ww_out,
                                float* __restrict__ sumww) {
    __shared__ float val[MM];
    __shared__ int   vix[MM];
    __shared__ float sp[MM];
    int b = blockIdx.x, t = threadIdx.x; // 1024 threads
    float u = DELTA + (1.f - DELTA) * usage_new[b * MM + t];
    val[t] = u; vix[t] = t;
    __syncthreads();
    // bitonic sort ascending on (value, index)  == jax.lax.top_k(-u) order
    for (int ksz = 2; ksz <= MM; ksz <<= 1) {
        for (int j = ksz >> 1; j > 0; j >>= 1) {
            int ixj = t ^ j;
            if (ixj > t) {
                bool up = ((t & ksz) == 0);
                float v1 = val[t], v2 = val[ixj];
                int i1 = vix[t], i2 = vix[ixj];
                bool sw = up ? (v1 > v2 || (v1 == v2 && i1 > i2))
                             : (v1 < v2 || (v1 == v2 && i1 < i2));
                if (sw) { val[t] = v2; val[ixj] = v1; vix[t] = i2; vix[ixj] = i1; }
            }
            __syncthreads();
        }
    }
    // inclusive product scan (Hillis-Steele)
    sp[t] = val[t]; __syncthreads();
    for (int o = 1; o < MM; o <<= 1) {
        float y = (t >= o) ? sp[t - o] : 1.f;
        __syncthreads();
        if (t >= o) sp[t] *= y;
        __syncthreads();
    }
    float ep = (t == 0) ? 1.f : sp[t - 1];          // exclusive product
    float sa = (1.f - val[t]) * ep;                  // sorted allocation
    __syncthreads();
    sp[t] = sa;
    __syncthreads();
    float alloc_t = sp[vix[t]];                      // reference's gather via phi
    float ag = ws[OFF_AG + b], wg = ws[OFF_WG + b];
    float w = wg * (ag * alloc_t + (1.f - ag) * wcw[b * MM + t]);
    ww_out[b * MM + t] = w;
    __syncthreads();
    val[t] = w; __syncthreads();
    for (int o = 512; o > 0; o >>= 1) { if (t < o) val[t] += val[t + o]; __syncthreads(); }
    if (t == 0) sumww[b] = val[0];
}

// ============ memory update (streaming float4, 512K vec4 elems) ============
__global__ void mem_update(const float* __restrict__ memory,
                           const float* __restrict__ ww,
                           const float* __restrict__ ws,
                           float* __restrict__ out_mem) {
    int i4 = blockIdx.x * blockDim.x + threadIdx.x; // 0..524287
    int b = i4 >> 14, m = (i4 >> 4) & (MM - 1), j4 = (i4 & 15) << 2;
    float w = ww[b * MM + m];
    v4f e = *(const v4f*)(ws + OFF_EV + b * WW_ + j4);
    v4f v = *(const v4f*)(ws + OFF_WV + b * WW_ + j4);
    v4f mm = __builtin_nontemporal_load((const v4f*)(memory + (size_t)i4 * 4));
    v4f o;
    o.x = mm.x * (1.f - w * e.x) + w * v.x;
    o.y = mm.y * (1.f - w * e.y) + w * v.y;
    o.z = mm.z * (1.f - w * e.z) + w * v.z;
    o.w = mm.w * (1.f - w * e.w) + w * v.w;
    __builtin_nontemporal_store(o, (v4f*)(out_mem + (size_t)i4 * 4));
}

// ============ link matrix (dominant HBM stream, float4 non-temporal) ============
__global__ void link_kernel(const float* __restrict__ link,
                            const float* __restrict__ ww,
                            const float* __restrict__ prec,
                            float* __restrict__ out) {
    long tid = (long)blockIdx.x * blockDim.x + threadIdx.x; // 8,388,608 float4s
    int b  = (int)(tid >> 18);
    int i  = (int)(tid >> 8) & (MM - 1);
    int j4 = ((int)tid & 255) << 2;
    const float* wwb = ww + b * MM;
    float wi = wwb[i];
    size_t base = ((size_t)b * MM + i) * MM + j4;
    v4f l  = __builtin_nontemporal_load((const v4f*)(link + base));
    v4f wj = *(const v4f*)(wwb + j4);
    v4f p  = *(const v4f*)(prec + b * MM + j4);
    v4f o;
    o.x = (i == j4 + 0) ? 0.f : (1.f - wi - wj.x) * l.x + wi * p.x;
    o.y = (i == j4 + 1) ? 0.f : (1.f - wi - wj.y) * l.y + wi * p.y;
    o.z = (i == j4 + 2) ? 0.f : (1.f - wi - wj.z) * l.z + wi * p.z;
    o.w = (i == j4 + 3) ? 0.f : (1.f - wi - wj.w) * l.w + wi * p.w;
    __builtin_nontemporal_store(o, (v4f*)(out + base));
}

// ============ precedence update ============
__global__ void prec_kernel(const float* __restrict__ prec,
                            const float* __restrict__ ww,
                            const float* __restrict__ sumww,
                            float* __restrict__ out) {
    int i = blockIdx.x * blockDim.x + threadIdx.x; // 32768
    int b = i >> 10;
    out[i] = (1.f - sumww[b]) * prec[i] + ww[i];
}

// ============ d2 + top-8 selection + gather read vectors ============
__global__ void knn_kernel(const float* __restrict__ mem_new, // d_out O_MEM
                           const float* __restrict__ ws,
                           float* __restrict__ out_rv,
                           float* __restrict__ out_nrw) {
    __shared__ float d2s[MM];
    __shared__ float rmin[256];
    __shared__ int   rmix[256];
    __shared__ float sdist[KNN];
    __shared__ int   sidx[KNN];
    int b = blockIdx.x >> 2, r = blockIdx.x & 3, t = threadIdx.x;
    const v4f* rk4 = (const v4f*)(ws + OFF_RK + b * (RR * WW_) + r * WW_);
    float kn = 0.f;
    #pragma unroll
    for (int j = 0; j < WW_ / 4; ++j) {
        v4f k4 = rk4[j];
        kn += k4.x * k4.x + k4.y * k4.y + k4.z * k4.z + k4.w * k4.w;
    }
    for (int m = t; m < MM; m += 256) {
        const v4f* row4 = (const v4f*)(mem_new + (size_t)b * MM * WW_ + (size_t)m * WW_);
        float dot = 0.f, nn = 0.f;
        #pragma unroll
        for (int j = 0; j < WW_ / 4; ++j) {
            v4f r4 = row4[j], k4 = rk4[j];
            dot += r4.x * k4.x + r4.y * k4.y + r4.z * k4.z + r4.w * k4.w;
            nn  += r4.x * r4.x + r4.y * r4.y + r4.z * r4.z + r4.w * r4.w;
        }
        d2s[m] = nn - 2.f * dot + kn;
    }
    __syncthreads();
    for (int k = 0; k < KNN; ++k) {
        float mv = 3.4e38f; int mi = MM;
        for (int m = t; m < MM; m += 256) {
            float v = d2s[m];
            if (v < mv || (v == mv && m < mi)) { mv = v; mi = m; }
        }
        rmin[t] = mv; rmix[t] = mi; __syncthreads();
        for (int o = 128; o > 0; o >>= 1) {
            if (t < o) {
                if (rmin[t + o] < rmin[t] || (rmin[t + o] == rmin[t] && rmix[t + o] < rmix[t])) {
                    rmin[t] = rmin[t + o]; rmix[t] = rmix[t + o];
                }
            }
            __syncthreads();
        }
        if (t == 0) {
            sdist[k] = rmin[0];
            sidx[k]  = rmix[0];
            d2s[rmix[0]] = 3.4e38f;    // exclude from next pass
        }
        __syncthreads();
    }
    float dmax = sdist[KNN - 1];                      // ascending -> last is max
    if (t < KNN) out_nrw[(b * RR + r) * KNN + t] = sdist[t] / (dmax + DELTA);
    for (int kk = t >> 6; kk < KNN; kk += 4) {        // 256 threads copy 8x64 floats
        int j = t & (WW_ - 1);
        out_rv[((size_t)(b * RR + r) * KNN + kk) * WW_ + j] =
            mem_new[(size_t)b * MM * WW_ + (size_t)sidx[kk] * WW_ + j];
    }
}

extern "C" void kernel_launch(void* const* d_in, const int* in_sizes, int n_in,
                              void* d_out, int out_size, void* d_ws, size_t ws_size,
                              hipStream_t stream) {
    (void)in_sizes; (void)n_in; (void)out_size; (void)ws_size;
    const float* xi       = (const float*)d_in[0];
    const float* memory   = (const float*)d_in[1];
    const float* usage    = (const float*)d_in[2];
    const float* read_w   = (const float*)d_in[3];
    const float* write_w  = (const float*)d_in[4];
    const float* prec     = (const float*)d_in[5];
    const float* link     = (const float*)d_in[6];
    const float* W_rk = (const float*)d_in[7];  const float* b_rk = (const float*)d_in[8];
    const float* W_wk = (const float*)d_in[9];  const float* b_wk = (const float*)d_in[10];
    const float* W_ws = (const float*)d_in[11]; const float* b_ws = (const float*)d_in[12];
    const float* W_ev = (const float*)d_in[13]; const float* b_ev = (const float*)d_in[14];
    const float* W_wv = (const float*)d_in[15]; const float* b_wv = (const float*)d_in[16];
    const float* W_fg = (const float*)d_in[17]; const float* b_fg = (const float*)d_in[18];
    const float* W_ag = (const float*)d_in[19]; const float* b_ag = (const float*)d_in[20];
    const float* W_wg = (const float*)d_in[21]; const float* b_wg = (const float*)d_in[22];
    float* ws  = (float*)d_ws;
    float* out = (float*)d_out;

    // WMMA projections: read keys (N=256), write key / erase / write vector (N=64)
    gemm_wmma_f32<<<dim3(16, 2), 32, 0, stream>>>(xi, W_rk, b_rk, ws + OFF_RK, 256, 0);
    gemm_wmma_f32<<<dim3(4, 2),  32, 0, stream>>>(xi, W_wk, b_wk, ws + OFF_WK, 64, 0);
    gemm_wmma_f32<<<dim3(4, 2),  32, 0, stream>>>(xi, W_ev, b_ev, ws + OFF_EV, 64, 1);
    gemm_wmma_f32<<<dim3(4, 2),  32, 0, stream>>>(xi, W_wv, b_wv, ws + OFF_WV, 64, 0);
    small_proj<<<1, 224, 0, stream>>>(xi, W_ws, b_ws, W_fg, b_fg, W_ag, b_ag, W_wg, b_wg, ws);

    usage_kernel<<<128, 256, 0, stream>>>(usage, write_w, read_w, ws, out + O_USAGE);
    cos_softmax<<<BB, 256, 0, stream>>>(memory, ws, ws + OFF_WCW);
    alloc_ww_kernel<<<BB, 1024, 0, stream>>>(out + O_USAGE, ws + OFF_WCW, ws,
                                             out + O_WW, ws + OFF_SUMWW);
    mem_update<<<2048, 256, 0, stream>>>(memory, out + O_WW, ws, out + O_MEM);
    link_kernel<<<32768, 256, 0, stream>>>(link, out + O_WW, prec, out + O_LINK);
    prec_kernel<<<128, 256, 0, stream>>>(prec, out + O_WW, ws + OFF_SUMWW, out + O_PREC);
    knn_kernel<<<BB * RR, 256, 0, stream>>>(out + O_MEM, ws, out + O_RV, out + O_NRW);
}